// GNN_62199716381547
// MI455X (gfx1250) — compile-verified
//
#include <hip/hip_runtime.h>
#include <cstdint>

#define N_NODES 50000
#define N_EDGES 800000
#define N_TOT_EDGES (N_EDGES + N_NODES) /* 850000 incl. self loops */
#define F_IN 128
#define F_OUT 16

typedef __attribute__((ext_vector_type(2))) float v2f;
typedef __attribute__((ext_vector_type(8))) float v8f;

// ---------------- small utility kernels ----------------

__global__ void __launch_bounds__(256) zero_f32_kernel(float* __restrict__ p, int n4) {
  int i = blockIdx.x * blockDim.x + threadIdx.x;
  if (i < n4) ((float4*)p)[i] = make_float4(0.f, 0.f, 0.f, 0.f);
}

__global__ void __launch_bounds__(256) init_deg_kernel(float* __restrict__ deg) {
  int i = blockIdx.x * blockDim.x + threadIdx.x;
  if (i < N_NODES) deg[i] = 1.0f;  // self loop contribution
}

__global__ void __launch_bounds__(256) count_deg_kernel(const long long* __restrict__ col,
                                                        float* __restrict__ deg) {
  int e = blockIdx.x * blockDim.x + threadIdx.x;
  if (e < N_EDGES) atomicAdd(&deg[(int)col[e]], 1.0f);
}

__global__ void __launch_bounds__(256) dinv_kernel(const float* __restrict__ deg,
                                                   float* __restrict__ dinv) {
  int i = blockIdx.x * blockDim.x + threadIdx.x;
  if (i < N_NODES) {
    float d = deg[i];
    dinv[i] = d > 0.f ? rsqrtf(d) : 0.f;
  }
}

// ---------------- GEMM1: h1 = x @ W1 via fp32 WMMA ----------------
// One wave per 16-row tile; 8 accumulator tiles cover all 128 output cols.
// A 16x4 fp32 layout: lanes 0-15 hold K=k0,k0+1 ; lanes 16-31 hold K=k0+2,k0+3.
__global__ void __launch_bounds__(32) gemm1_wmma_kernel(const float* __restrict__ x,
                                                        const float* __restrict__ W1,
                                                        float* __restrict__ h1) {
  const int mtile = blockIdx.x;         // 0..3124
  const int lane  = threadIdx.x;        // 0..31
  const int nlane = lane & 15;
  const int khalf = lane >> 4;          // 0 -> K+{0,1}, 1 -> K+{2,3}

  const float* arow = x + (size_t)(mtile * 16 + nlane) * F_IN + 2 * khalf;

  v8f zero = {0.f, 0.f, 0.f, 0.f, 0.f, 0.f, 0.f, 0.f};
  v8f acc[8];
#pragma unroll
  for (int nt = 0; nt < 8; ++nt) acc[nt] = zero;

#pragma unroll 4
  for (int ks = 0; ks < 32; ++ks) {
    const int k0 = ks * 4;
    v2f a;
    a.x = arow[k0];
    a.y = arow[k0 + 1];
    const float* brow = W1 + (size_t)(k0 + 2 * khalf) * F_IN + nlane;
#pragma unroll
    for (int nt = 0; nt < 8; ++nt) {
      v2f b;
      b.x = brow[nt * 16];
      b.y = brow[F_IN + nt * 16];
      acc[nt] = __builtin_amdgcn_wmma_f32_16x16x4_f32(
          false, a, false, b, (short)0, acc[nt], false, false);
    }
  }

  // C/D layout: VGPR r -> M = r (lanes 0-15) or r+8 (lanes 16-31), N = lane&15
#pragma unroll
  for (int nt = 0; nt < 8; ++nt) {
#pragma unroll
    for (int r = 0; r < 8; ++r) {
      h1[(size_t)(mtile * 16 + r + 8 * khalf) * F_IN + nt * 16 + nlane] = acc[nt][r];
    }
  }
}

// ---------------- edge aggregation layer 1 (128 features) ----------------
// one wave per edge; float4 gather of h1[row], 4 fp32 atomics per lane into agg1[col]
__global__ void __launch_bounds__(256) scatter1_kernel(const long long* __restrict__ row,
                                                       const long long* __restrict__ col,
                                                       const float* __restrict__ dinv,
                                                       const float* __restrict__ h1,
                                                       float* __restrict__ agg1) {
  const int e    = blockIdx.x * 8 + (threadIdx.x >> 5);
  const int lane = threadIdx.x & 31;
  if (e >= N_TOT_EDGES) return;
  int r, c;
  if (e < N_EDGES) {
    r = (int)row[e];
    c = (int)col[e];
  } else {
    r = c = e - N_EDGES;  // self loop
  }
  const float nrm = dinv[r] * dinv[c];
  const float4 v = ((const float4*)(h1 + (size_t)r * F_IN))[lane];
  float* dst = agg1 + (size_t)c * F_IN + lane * 4;
  atomicAdd(dst + 0, v.x * nrm);
  atomicAdd(dst + 1, v.y * nrm);
  atomicAdd(dst + 2, v.z * nrm);
  atomicAdd(dst + 3, v.w * nrm);
}

// ---------------- GEMM2: h2 = relu(agg1 + b1) @ W2 via fp32 WMMA ----------------
__global__ void __launch_bounds__(32) gemm2_wmma_kernel(const float* __restrict__ agg1,
                                                        const float* __restrict__ b1,
                                                        const float* __restrict__ W2,
                                                        float* __restrict__ h2) {
  const int mtile = blockIdx.x;
  const int lane  = threadIdx.x;
  const int nlane = lane & 15;
  const int khalf = lane >> 4;

  const float* arow = agg1 + (size_t)(mtile * 16 + nlane) * F_IN + 2 * khalf;

  v8f acc = {0.f, 0.f, 0.f, 0.f, 0.f, 0.f, 0.f, 0.f};

#pragma unroll 4
  for (int ks = 0; ks < 32; ++ks) {
    const int k0 = ks * 4;
    const int ka = k0 + 2 * khalf;
    v2f a;
    a.x = fmaxf(arow[k0] + b1[ka], 0.f);       // fused bias + relu
    a.y = fmaxf(arow[k0 + 1] + b1[ka + 1], 0.f);
    v2f b;
    b.x = W2[(size_t)ka * F_OUT + nlane];
    b.y = W2[(size_t)(ka + 1) * F_OUT + nlane];
    acc = __builtin_amdgcn_wmma_f32_16x16x4_f32(
        false, a, false, b, (short)0, acc, false, false);
  }

#pragma unroll
  for (int r = 0; r < 8; ++r) {
    h2[(size_t)(mtile * 16 + r + 8 * khalf) * F_OUT + nlane] = acc[r];
  }
}

// ---------------- edge aggregation layer 2 (16 features) ----------------
__global__ void __launch_bounds__(256) scatter2_kernel(const long long* __restrict__ row,
                                                       const long long* __restrict__ col,
                                                       const float* __restrict__ dinv,
                                                       const float* __restrict__ h2,
                                                       float* __restrict__ out) {
  const int idx = blockIdx.x * 256 + threadIdx.x;
  const int e = idx >> 4;
  const int f = idx & 15;
  if (e >= N_TOT_EDGES) return;
  int r, c;
  if (e < N_EDGES) {
    r = (int)row[e];
    c = (int)col[e];
  } else {
    r = c = e - N_EDGES;
  }
  const float nrm = dinv[r] * dinv[c];
  atomicAdd(&out[(size_t)c * F_OUT + f], h2[(size_t)r * F_OUT + f] * nrm);
}

// ---------------- bias + log_softmax (in place on d_out) ----------------
__global__ void __launch_bounds__(256) finalize_kernel(float* __restrict__ out,
                                                       const float* __restrict__ b2) {
  int i = blockIdx.x * blockDim.x + threadIdx.x;
  if (i >= N_NODES) return;
  float v[F_OUT];
  float m = -3.0e38f;
#pragma unroll
  for (int j = 0; j < F_OUT; ++j) {
    v[j] = out[(size_t)i * F_OUT + j] + b2[j];
    m = fmaxf(m, v[j]);
  }
  float s = 0.f;
#pragma unroll
  for (int j = 0; j < F_OUT; ++j) s += expf(v[j] - m);
  const float lse = m + logf(s);
#pragma unroll
  for (int j = 0; j < F_OUT; ++j) out[(size_t)i * F_OUT + j] = v[j] - lse;
}

extern "C" void kernel_launch(void* const* d_in, const int* in_sizes, int n_in,
                              void* d_out, int out_size, void* d_ws, size_t ws_size,
                              hipStream_t stream) {
  const float*     x  = (const float*)d_in[0];
  const long long* ei = (const long long*)d_in[1];  // int64 edge_index [2, 800000]
  const float*     W1 = (const float*)d_in[2];
  const float*     b1 = (const float*)d_in[3];
  const float*     W2 = (const float*)d_in[4];
  const float*     b2 = (const float*)d_in[5];
  const long long* row = ei;             // sources
  const long long* col = ei + N_EDGES;   // destinations
  float* out = (float*)d_out;

  // workspace carve-up (floats), 256B-aligned chunks
  float* deg  = (float*)d_ws;                         // 50176
  float* dinv = deg + 50176;                          // 50176
  float* h1   = dinv + 50176;                         // 6,400,000
  float* agg1 = h1 + (size_t)N_NODES * F_IN;          // 6,400,000
  float* h2   = agg1 + (size_t)N_NODES * F_IN;        // 800,000

  init_deg_kernel<<<(N_NODES + 255) / 256, 256, 0, stream>>>(deg);
  count_deg_kernel<<<N_EDGES / 256, 256, 0, stream>>>(col, deg);
  dinv_kernel<<<(N_NODES + 255) / 256, 256, 0, stream>>>(deg, dinv);

  gemm1_wmma_kernel<<<N_NODES / 16, 32, 0, stream>>>(x, W1, h1);

  zero_f32_kernel<<<(N_NODES * F_IN / 4 + 255) / 256, 256, 0, stream>>>(agg1,
                                                                        N_NODES * F_IN / 4);
  scatter1_kernel<<<(N_TOT_EDGES + 7) / 8, 256, 0, stream>>>(row, col, dinv, h1, agg1);

  gemm2_wmma_kernel<<<N_NODES / 16, 32, 0, stream>>>(agg1, b1, W2, h2);

  zero_f32_kernel<<<(N_NODES * F_OUT / 4 + 255) / 256, 256, 0, stream>>>(out,
                                                                         N_NODES * F_OUT / 4);
  scatter2_kernel<<<(N_TOT_EDGES * 16 + 255) / 256, 256, 0, stream>>>(row, col, dinv, h2, out);

  finalize_kernel<<<(N_NODES + 255) / 256, 256, 0, stream>>>(out, b2);
}